// BodyOrderedModel_15358803050681
// MI455X (gfx1250) — compile-verified
//
#include <hip/hip_runtime.h>
#include <hip/hip_bf16.h>

// ---------------------------------------------------------------------------
// MACE-like BodyOrderedModel forward + forces, fp32, CDNA5 (gfx1250).
// All 64x64 channel-mixing GEMMs run on V_WMMA_F32_16X16X4_F32.
// Weight tiles are staged in LDS, swizzled so every B fragment is one
// aligned ds_load_b64.
// ---------------------------------------------------------------------------

typedef __attribute__((ext_vector_type(2))) float f32x2;
typedef __attribute__((ext_vector_type(8))) float f32x8;

#define NN 8000
#define EE 128000
#define GG 16
#define KK 64
#define TT 2
#define NBB 8
#define NEE 10
#define LM 16

// ---------------------------------------------------------------------------
// LDS weight staging.
//  SWIZ=true  (for C = A@W):   sW[(k>>1)*128 + 2n + (k&1)] = W[k*64+n]
//     -> B frag {W[kk][n], W[kk+1][n]} is one aligned 8-byte LDS load.
//  SWIZ=false (for C = A@W^T): plain row-major copy; frag {W[n][kk],W[n][kk+1]}
//     is already contiguous.
// ---------------------------------------------------------------------------
template <bool SWIZ>
__device__ __forceinline__ void lds_fill(float* __restrict__ sW,
                                         const float* __restrict__ W) {
  for (int i = threadIdx.x; i < 4096; i += blockDim.x) {
    if (SWIZ) {
      const int k = i >> 6, n = i & 63;
      sW[(k >> 1) * 128 + n * 2 + (k & 1)] = W[i];
    } else {
      sW[i] = W[i];
    }
  }
}

// ---------------------------------------------------------------------------
// Wave-level 16x64 GEMM tile: acc += A[16 x 64] * W[64 x 64] (W in LDS).
// f32 WMMA 16x16x4 layout:
//   A frag: lane m = lane&15 -> row, kh = lane>>4 selects K pair {2kh, 2kh+1}
//   B frag: lane n = lane&15 -> col, same K pair
//   C/D:    vgpr i: row = i + 8*(lane>>4), col = lane&15 (per 16-col tile)
// ---------------------------------------------------------------------------
template <bool TRANSW>
__device__ __forceinline__ void wave_gemm64(const float* __restrict__ A, int sA,
                                            const float* __restrict__ sW,
                                            f32x8 acc[4]) {
  const int lane = threadIdx.x & 31;
  const int m = lane & 15;
  const int kh = lane >> 4;
  const float* arow = A + (size_t)m * sA;
#pragma unroll 4
  for (int kb = 0; kb < 16; ++kb) {
    const int kk = kb * 4 + kh * 2;
    const f32x2 a = *(const f32x2*)(arow + kk);
#pragma unroll
    for (int nt = 0; nt < 4; ++nt) {
      const int n = nt * 16 + m;
      f32x2 b;
      if (TRANSW) {
        b = *(const f32x2*)(sW + n * 64 + kk);  // plain layout
      } else {
        b = *(const f32x2*)(sW + (kk >> 1) * 128 + 2 * n);  // swizzled layout
      }
      acc[nt] = __builtin_amdgcn_wmma_f32_16x16x4_f32(
          false, a, false, b, (short)0, acc[nt], false, false);
    }
  }
}

__device__ __forceinline__ void wave_acc_zero(f32x8 acc[4]) {
#pragma unroll
  for (int nt = 0; nt < 4; ++nt)
#pragma unroll
    for (int i = 0; i < 8; ++i) acc[nt][i] = 0.0f;
}

template <bool ADD>
__device__ __forceinline__ void wave_store64(const f32x8 acc[4],
                                             float* __restrict__ C, int sC) {
  const int lane = threadIdx.x & 31;
  const int n0 = lane & 15;
  const int mh = (lane >> 4) * 8;
#pragma unroll
  for (int nt = 0; nt < 4; ++nt)
#pragma unroll
    for (int i = 0; i < 8; ++i) {
      const size_t idx = (size_t)(mh + i) * sC + nt * 16 + n0;
      C[idx] = (ADD ? C[idx] : 0.0f) + acc[nt][i];
    }
}

// Generic C[M x 64] = (ADD? C : 0) + A[M x 64] @ W(64x64, opt transposed)
template <bool TRANSW, bool ADD>
__global__ void gemm64_kernel(const float* __restrict__ A, int sA,
                              const float* __restrict__ W,
                              float* __restrict__ C, int sC, int M) {
  __shared__ float sW[4096];
  lds_fill<!TRANSW>(sW, W);
  __syncthreads();
  const int wave = threadIdx.x >> 5;
  const int rowBase = blockIdx.x * 128 + wave * 16;
  if (rowBase >= M) return;
  const float* Ar = A + (size_t)rowBase * sA;
  __builtin_prefetch(Ar, 0, 0);
  f32x8 acc[4];
  wave_acc_zero(acc);
  wave_gemm64<TRANSW>(Ar, sA, sW, acc);
  wave_store64<ADD>(acc, C + (size_t)rowBase * sC, sC);
}

__device__ __forceinline__ int l_weight(int l) {
  return (l == 0) ? 0 : (l < 4) ? 1 : (l < 9) ? 2 : 3;
}

// nf_out[:,l,:] = agg[:,l,:] @ O_{L(l)} + nf_in[:,l,:] @ S_{L(l)}
__global__ void mix_fwd_kernel(const float* __restrict__ agg,
                               const float* __restrict__ nf_in,
                               const float* __restrict__ Wout_t,
                               const float* __restrict__ Wskip_t,
                               float* __restrict__ nf_out, int M) {
  __shared__ float sO[4096];
  __shared__ float sS[4096];
  const int l = blockIdx.y;
  const int lw = l_weight(l);
  lds_fill<true>(sO, Wout_t + (size_t)lw * 4096);
  lds_fill<true>(sS, Wskip_t + (size_t)lw * 4096);
  __syncthreads();
  const int wave = threadIdx.x >> 5;
  const int rowBase = blockIdx.x * 128 + wave * 16;
  if (rowBase >= M) return;
  const size_t rofs = (size_t)rowBase * 1024 + (size_t)l * 64;
  f32x8 acc[4];
  wave_acc_zero(acc);
  wave_gemm64<false>(agg + rofs, 1024, sO, acc);
  wave_gemm64<false>(nf_in + rofs, 1024, sS, acc);
  wave_store64<false>(acc, nf_out + rofs, 1024);
}

// gagg[:,l,:] = gnf[:,l,:] @ O^T ; gnf_prev[:,l,:] = gnf[:,l,:] @ S^T
__global__ void mix_bwd_kernel(const float* __restrict__ gnf,
                               const float* __restrict__ Wout_t,
                               const float* __restrict__ Wskip_t,
                               float* __restrict__ gagg,
                               float* __restrict__ gnf_prev, int M) {
  __shared__ float sO[4096];
  __shared__ float sS[4096];
  const int l = blockIdx.y;
  const int lw = l_weight(l);
  lds_fill<false>(sO, Wout_t + (size_t)lw * 4096);
  lds_fill<false>(sS, Wskip_t + (size_t)lw * 4096);
  __syncthreads();
  const int wave = threadIdx.x >> 5;
  const int rowBase = blockIdx.x * 128 + wave * 16;
  if (rowBase >= M) return;
  const size_t rofs = (size_t)rowBase * 1024 + (size_t)l * 64;
  f32x8 acc1[4], acc2[4];
  wave_acc_zero(acc1);
  wave_acc_zero(acc2);
  wave_gemm64<true>(gnf + rofs, 1024, sO, acc1);
  wave_gemm64<true>(gnf + rofs, 1024, sS, acc2);
  wave_store64<false>(acc1, gagg + rofs, 1024);
  wave_store64<false>(acc2, gnf_prev + rofs, 1024);
}

// ---------------------------------------------------------------------------
// Elementwise / scatter kernels
// ---------------------------------------------------------------------------
__global__ void zero_kernel(float* __restrict__ p, size_t n) {
  size_t i = (size_t)blockIdx.x * blockDim.x + threadIdx.x;
  size_t stride = (size_t)gridDim.x * blockDim.x;
  for (; i < n; i += stride) p[i] = 0.0f;
}

#define S3 1.7320508075688772f
#define S5 2.23606797749979f
#define S7 2.6457513110645907f
#define S15 3.872983346207417f
#define S42 6.48074069840786f
#define S70 8.366600265340756f
#define S105 10.246950765959598f
#define PI_F 3.14159265358979323846f

__global__ void edge_geom_kernel(const float* __restrict__ pos,
                                 const float* __restrict__ shifts,
                                 const int* __restrict__ send,
                                 const int* __restrict__ recv,
                                 float* __restrict__ vec, float* __restrict__ rr,
                                 float* __restrict__ sh, float* __restrict__ ef) {
  const int e = blockIdx.x * blockDim.x + threadIdx.x;
  if (e >= EE) return;
  const int sd = send[e], rc = recv[e];
  const float vx = pos[rc * 3 + 0] - pos[sd * 3 + 0] + shifts[e * 3 + 0];
  const float vy = pos[rc * 3 + 1] - pos[sd * 3 + 1] + shifts[e * 3 + 1];
  const float vz = pos[rc * 3 + 2] - pos[sd * 3 + 2] + shifts[e * 3 + 2];
  const float r = sqrtf(vx * vx + vy * vy + vz * vz);
  vec[e * 3 + 0] = vx;
  vec[e * 3 + 1] = vy;
  vec[e * 3 + 2] = vz;
  rr[e] = r;
  const float inv = 1.0f / (r + 1e-9f);
  const float x = vx * inv, y = vy * inv, z = vz * inv;
  const float x2 = x * x, y2 = y * y, z2 = z * z;
  float* s = sh + (size_t)e * 16;
  s[0] = 1.0f;
  s[1] = S3 * y;
  s[2] = S3 * z;
  s[3] = S3 * x;
  s[4] = S15 * x * y;
  s[5] = S15 * y * z;
  s[6] = 0.5f * S5 * (3.0f * z2 - 1.0f);
  s[7] = S15 * x * z;
  s[8] = 0.5f * S15 * (x2 - y2);
  s[9] = 0.25f * S70 * y * (3.0f * x2 - y2);
  s[10] = S105 * x * y * z;
  s[11] = 0.25f * S42 * y * (5.0f * z2 - 1.0f);
  s[12] = 0.5f * S7 * z * (5.0f * z2 - 3.0f);
  s[13] = 0.25f * S42 * x * (5.0f * z2 - 1.0f);
  s[14] = 0.5f * S105 * z * (x2 - y2);
  s[15] = 0.25f * S70 * x * (x2 - y2);
  // radial: bessel * polynomial cutoff (R_MAX=5, P_CUT=6)
  const float rp = fmaxf(r, 1e-9f);
  const float u = r * 0.2f;
  float fc = 0.0f;
  if (u < 1.0f) {
    const float u2 = u * u, u3 = u2 * u, u6 = u3 * u3;
    fc = 1.0f - 28.0f * u6 + 48.0f * u6 * u - 21.0f * u6 * u2;
  }
  const float Cb = 0.6324555320336759f;  // sqrt(2/5)
  const float irp = 1.0f / rp;
#pragma unroll
  for (int b = 0; b < NBB; ++b) {
    const float an = (float)(b + 1) * (PI_F / 5.0f);
    ef[(size_t)e * 8 + b] = Cb * sinf(an * rp) * irp * fc;
  }
}

__global__ void embed_e0_kernel(const int* __restrict__ node_z,
                                const int* __restrict__ batch,
                                const float* __restrict__ ae,
                                const float* __restrict__ W_emb,
                                float* __restrict__ nf0,
                                float* __restrict__ contrib) {
  const int idx = blockIdx.x * blockDim.x + threadIdx.x;
  if (idx >= NN * 64) return;
  const int n = idx >> 6, k = idx & 63;
  const int z = node_z[n];
  float* row = nf0 + (size_t)n * 1024;
  row[k] = W_emb[z * 64 + k];
#pragma unroll
  for (int l = 1; l < LM; ++l) row[l * 64 + k] = 0.0f;
  if (k == 0) atomicAdd(&contrib[batch[n] * 3 + 0], ae[z]);
}

__global__ void rad1_silu_kernel(const float* __restrict__ ef,
                                 const float* __restrict__ R1,  // (8,64)
                                 float* __restrict__ pre,
                                 float* __restrict__ sbuf) {
  const int idx = blockIdx.x * blockDim.x + threadIdx.x;
  if (idx >= EE * 64) return;
  const int e = idx >> 6, k = idx & 63;
  const float* efe = ef + (size_t)e * 8;
  float p = 0.0f;
#pragma unroll
  for (int b = 0; b < NBB; ++b) p += efe[b] * R1[b * 64 + k];
  pre[idx] = p;
  sbuf[idx] = p / (1.0f + expf(-p));
}

__global__ void qscatter_kernel(const float* __restrict__ rad,
                                const float* __restrict__ h,
                                const float* __restrict__ sh,
                                const int* __restrict__ send,
                                const int* __restrict__ recv,
                                float* __restrict__ q,
                                float* __restrict__ agg) {
  const int idx = blockIdx.x * blockDim.x + threadIdx.x;
  if (idx >= EE * 64) return;
  const int e = idx >> 6, k = idx & 63;
  const float qv = rad[idx] * h[(size_t)send[e] * 64 + k];
  q[idx] = qv;
  const float* shr = sh + (size_t)e * 16;
  float* a = agg + (size_t)recv[e] * 1024 + k;
#pragma unroll
  for (int l = 0; l < LM; ++l) atomicAdd(&a[l * 64], shr[l] * qv);
}

__global__ void readout_kernel(const float* __restrict__ nf,
                               const float* __restrict__ Wr,
                               const int* __restrict__ batch,
                               float* __restrict__ contrib, int slot) {
  const int n = blockIdx.x * blockDim.x + threadIdx.x;
  if (n >= NN) return;
  const float* row = nf + (size_t)n * 1024;
  float s = 0.0f;
#pragma unroll 8
  for (int k = 0; k < KK; ++k) s += row[k] * Wr[k];
  atomicAdd(&contrib[batch[n] * 3 + slot], s);
}

__global__ void energy_kernel(const float* __restrict__ contrib,
                              float* __restrict__ energy) {
  const int g = blockIdx.x * blockDim.x + threadIdx.x;
  if (g >= GG) return;
  energy[g] = contrib[g * 3 + 0] + contrib[g * 3 + 1] + contrib[g * 3 + 2];
}

__global__ void gnf_init_kernel(const float* __restrict__ Wr1,
                                float* __restrict__ gnf) {
  const size_t idx = (size_t)blockIdx.x * blockDim.x + threadIdx.x;
  if (idx >= (size_t)NN * 1024) return;
  const int k = idx & 63, l = (idx >> 6) & 15;
  gnf[idx] = (l == 0) ? Wr1[k] : 0.0f;
}

__global__ void addread_kernel(const float* __restrict__ Wr0,
                               float* __restrict__ gnf) {
  const int idx = blockIdx.x * blockDim.x + threadIdx.x;
  if (idx >= NN * 64) return;
  const int n = idx >> 6, k = idx & 63;
  gnf[(size_t)n * 1024 + k] += Wr0[k];
}

__global__ void gsh_kernel(const float* __restrict__ gagg,
                           const float* __restrict__ q,
                           const int* __restrict__ recv,
                           float* __restrict__ gsh) {
  const int idx = blockIdx.x * blockDim.x + threadIdx.x;
  if (idx >= EE * 16) return;
  const int e = idx >> 4, l = idx & 15;
  const float* ga = gagg + (size_t)recv[e] * 1024 + l * 64;
  const float* qe = q + (size_t)e * 64;
  float s = 0.0f;
#pragma unroll 8
  for (int k = 0; k < KK; ++k) s += ga[k] * qe[k];
  gsh[idx] += s;
}

__global__ void gq_kernel(const float* __restrict__ gagg,
                          const float* __restrict__ sh,
                          const float* __restrict__ h,
                          const float* __restrict__ rad,
                          const int* __restrict__ send,
                          const int* __restrict__ recv,
                          float* __restrict__ grad_rad,
                          float* __restrict__ gh) {
  const int idx = blockIdx.x * blockDim.x + threadIdx.x;
  if (idx >= EE * 64) return;
  const int e = idx >> 6, k = idx & 63;
  const int rc = recv[e], sd = send[e];
  const float* ga = gagg + (size_t)rc * 1024 + k;
  const float* shr = sh + (size_t)e * 16;
  float g = 0.0f;
#pragma unroll
  for (int l = 0; l < LM; ++l) g += ga[l * 64] * shr[l];
  grad_rad[idx] = g * h[(size_t)sd * 64 + k];
  atomicAdd(&gh[(size_t)sd * 64 + k], g * rad[idx]);
}

__global__ void gpre_kernel(const float* __restrict__ pre,
                            float* __restrict__ gs) {
  const int idx = blockIdx.x * blockDim.x + threadIdx.x;
  if (idx >= EE * 64) return;
  const float x = pre[idx];
  const float sig = 1.0f / (1.0f + expf(-x));
  gs[idx] *= sig * (1.0f + x * (1.0f - sig));
}

__global__ void gef_kernel(const float* __restrict__ gpre,
                           const float* __restrict__ R1,  // (8,64)
                           float* __restrict__ gef) {
  const int idx = blockIdx.x * blockDim.x + threadIdx.x;
  if (idx >= EE * 8) return;
  const int e = idx >> 3, b = idx & 7;
  const float* gp = gpre + (size_t)e * 64;
  const float* r1 = R1 + b * 64;
  float s = 0.0f;
#pragma unroll 8
  for (int k = 0; k < KK; ++k) s += gp[k] * r1[k];
  gef[idx] += s;
}

__global__ void force_kernel(const float* __restrict__ gsh,
                             const float* __restrict__ gef,
                             const float* __restrict__ vec,
                             const float* __restrict__ rr,
                             const int* __restrict__ send,
                             const int* __restrict__ recv,
                             float* __restrict__ forces) {
  const int e = blockIdx.x * blockDim.x + threadIdx.x;
  if (e >= EE) return;
  const float vx = vec[e * 3 + 0], vy = vec[e * 3 + 1], vz = vec[e * 3 + 2];
  const float r = rr[e];
  const float inv = 1.0f / (r + 1e-9f);
  const float x = vx * inv, y = vy * inv, z = vz * inv;
  const float x2 = x * x, y2 = y * y, z2 = z * z;
  const float* G = gsh + (size_t)e * 16;
  // dL/du from spherical harmonics
  float gx = S3 * G[3] + S15 * y * G[4] + S15 * z * G[7] + S15 * x * G[8] +
             1.5f * S70 * x * y * G[9] + S105 * y * z * G[10] +
             0.25f * S42 * (5.0f * z2 - 1.0f) * G[13] + S105 * x * z * G[14] +
             0.25f * S70 * (3.0f * x2 - y2) * G[15];
  float gy = S3 * G[1] + S15 * x * G[4] + S15 * z * G[5] - S15 * y * G[8] +
             0.75f * S70 * (x2 - y2) * G[9] + S105 * x * z * G[10] +
             0.25f * S42 * (5.0f * z2 - 1.0f) * G[11] - S105 * y * z * G[14] -
             0.5f * S70 * x * y * G[15];
  float gz = S3 * G[2] + S15 * y * G[5] + 3.0f * S5 * z * G[6] +
             S15 * x * G[7] + S105 * x * y * G[10] +
             2.5f * S42 * y * z * G[11] + S7 * (7.5f * z2 - 1.5f) * G[12] +
             2.5f * S42 * x * z * G[13] + 0.5f * S105 * (x2 - y2) * G[14];
  // du/dvec: u = vec/(r+eps)
  const float invr = (r > 0.0f) ? 1.0f / r : 0.0f;
  const float dotg = gx * vx + gy * vy + gz * vz;
  const float c2 = dotg * invr * inv * inv;
  float gvx = inv * gx - c2 * vx;
  float gvy = inv * gy - c2 * vy;
  float gvz = inv * gz - c2 * vz;
  // radial chain: d(ef_b)/dr
  const float rp = fmaxf(r, 1e-9f);
  const float irp = 1.0f / rp;
  const float u = r * 0.2f;
  float fc = 0.0f, dfc = 0.0f;
  if (u < 1.0f) {
    const float u2 = u * u, u3 = u2 * u, u5 = u2 * u3, u6 = u3 * u3;
    fc = 1.0f - 28.0f * u6 + 48.0f * u6 * u - 21.0f * u6 * u2;
    const float om = 1.0f - u;
    dfc = 0.2f * (-168.0f * u5 * om * om);
  }
  const float Cb = 0.6324555320336759f;
  const float* Ge = gef + (size_t)e * 8;
  float gr = 0.0f;
  if (r > 1e-9f) {
#pragma unroll
    for (int b = 0; b < NBB; ++b) {
      const float an = (float)(b + 1) * (PI_F / 5.0f);
      const float sn = sinf(an * rp), cn = cosf(an * rp);
      const float bes = Cb * sn * irp;
      const float dbes = Cb * (an * cn * irp - sn * irp * irp);
      gr += Ge[b] * (dbes * fc + bes * dfc);
    }
  }
  gvx += gr * vx * invr;
  gvy += gr * vy * invr;
  gvz += gr * vz * invr;
  // vec = pos[recv] - pos[send]; forces = -dL/dpos
  const int sd = send[e], rc = recv[e];
  atomicAdd(&forces[rc * 3 + 0], -gvx);
  atomicAdd(&forces[rc * 3 + 1], -gvy);
  atomicAdd(&forces[rc * 3 + 2], -gvz);
  atomicAdd(&forces[sd * 3 + 0], gvx);
  atomicAdd(&forces[sd * 3 + 1], gvy);
  atomicAdd(&forces[sd * 3 + 2], gvz);
}

// ---------------------------------------------------------------------------
// Host-side orchestration
// ---------------------------------------------------------------------------
static inline int cdiv(long a, long b) { return (int)((a + b - 1) / b); }

extern "C" void kernel_launch(void* const* d_in, const int* in_sizes, int n_in,
                              void* d_out, int out_size, void* d_ws,
                              size_t ws_size, hipStream_t stream) {
  const float* positions = (const float*)d_in[0];
  const float* shifts = (const float*)d_in[1];
  const float* atomic_energies = (const float*)d_in[2];
  const float* W_emb = (const float*)d_in[3];
  const float* W_up = (const float*)d_in[4];    // (T,4,64,64)
  const float* W_rad1 = (const float*)d_in[5];  // (T,8,64)
  const float* W_rad2 = (const float*)d_in[6];  // (T,64,64)
  const float* W_out = (const float*)d_in[7];   // (T,4,64,64)
  const float* W_skip = (const float*)d_in[8];  // (T,4,64,64)
  const float* W_read = (const float*)d_in[9];  // (T,64)
  const int* node_z = (const int*)d_in[10];
  const int* batch = (const int*)d_in[11];
  const int* edge_index = (const int*)d_in[12];
  (void)d_in[13];
  (void)in_sizes;
  (void)n_in;
  (void)ws_size;

  const int* send = edge_index;
  const int* recv = edge_index + EE;

  float* outp = (float*)d_out;
  float* energy = outp;        // 16
  float* contrib = outp + 16;  // 16*3
  float* forces = outp + 64;   // 8000*3

  // workspace carve-up
  float* w = (float*)d_ws;
  size_t off = 0;
  auto alloc = [&](size_t n) {
    float* p = w + off;
    off += n;
    return p;
  };
  float* vec = alloc((size_t)EE * 3);
  float* r = alloc((size_t)EE);
  float* sh = alloc((size_t)EE * 16);
  float* ef = alloc((size_t)EE * 8);
  float* nf[3];
  nf[0] = alloc((size_t)NN * 1024);
  nf[1] = alloc((size_t)NN * 1024);
  nf[2] = alloc((size_t)NN * 1024);
  float* h[2] = {alloc((size_t)NN * 64), alloc((size_t)NN * 64)};
  float* pre[2] = {alloc((size_t)EE * 64), alloc((size_t)EE * 64)};
  float* rad[2] = {alloc((size_t)EE * 64), alloc((size_t)EE * 64)};
  float* q[2] = {alloc((size_t)EE * 64), alloc((size_t)EE * 64)};
  float* sbuf = alloc((size_t)EE * 64);
  float* agg = alloc((size_t)NN * 1024);
  float* gnf_a = alloc((size_t)NN * 1024);
  float* gnf_b = alloc((size_t)NN * 1024);
  float* gagg = alloc((size_t)NN * 1024);
  float* gh = alloc((size_t)NN * 64);
  float* grad_rad = alloc((size_t)EE * 64);
  float* gs = alloc((size_t)EE * 64);
  float* gsh = alloc((size_t)EE * 16);
  float* gef = alloc((size_t)EE * 8);

  const int TB = 256;
  // zero outputs + accumulated grads
  zero_kernel<<<cdiv(out_size, TB), TB, 0, stream>>>(outp, (size_t)out_size);
  zero_kernel<<<cdiv((size_t)EE * 16, TB), TB, 0, stream>>>(gsh, (size_t)EE * 16);
  zero_kernel<<<cdiv((size_t)EE * 8, TB), TB, 0, stream>>>(gef, (size_t)EE * 8);

  // edge geometry: vec, r, spherical harmonics, radial basis
  edge_geom_kernel<<<cdiv(EE, TB), TB, 0, stream>>>(positions, shifts, send,
                                                    recv, vec, r, sh, ef);
  // node embedding + E0
  embed_e0_kernel<<<cdiv((size_t)NN * 64, TB), TB, 0, stream>>>(
      node_z, batch, atomic_energies, W_emb, nf[0], contrib);

  // ---------------- forward layers ----------------
  for (int t = 0; t < TT; ++t) {
    const float* U = W_up + (size_t)t * 4 * 4096;  // l=0 block only
    const float* R1 = W_rad1 + (size_t)t * 512;
    const float* R2 = W_rad2 + (size_t)t * 4096;
    const float* O = W_out + (size_t)t * 4 * 4096;
    const float* S = W_skip + (size_t)t * 4 * 4096;

    // h = nf[:,0,:] @ U0
    gemm64_kernel<false, false><<<cdiv(NN, 128), TB, 0, stream>>>(
        nf[t], 1024, U, h[t], 64, NN);
    // pre = ef @ R1 ; s = silu(pre)
    rad1_silu_kernel<<<cdiv((size_t)EE * 64, TB), TB, 0, stream>>>(ef, R1,
                                                                   pre[t], sbuf);
    // rad = s @ R2
    gemm64_kernel<false, false><<<cdiv(EE, 128), TB, 0, stream>>>(
        sbuf, 64, R2, rad[t], 64, EE);
    // agg = scatter_add(sh ⊗ (rad * h[send]))
    zero_kernel<<<cdiv((size_t)NN * 1024, TB), TB, 0, stream>>>(
        agg, (size_t)NN * 1024);
    qscatter_kernel<<<cdiv((size_t)EE * 64, TB), TB, 0, stream>>>(
        rad[t], h[t], sh, send, recv, q[t], agg);
    // nf_{t+1} = agg @ O_{L(l)} + nf_t @ S_{L(l)}
    dim3 mg(cdiv(NN, 128), 16);
    mix_fwd_kernel<<<mg, TB, 0, stream>>>(agg, nf[t], O, S, nf[t + 1], NN);
    // readout
    readout_kernel<<<cdiv(NN, TB), TB, 0, stream>>>(nf[t + 1], W_read + t * 64,
                                                    batch, contrib, t + 1);
  }
  energy_kernel<<<1, 32, 0, stream>>>(contrib, energy);

  // ---------------- backward (forces) ----------------
  gnf_init_kernel<<<cdiv((size_t)NN * 1024, TB), TB, 0, stream>>>(W_read + 64,
                                                                  gnf_a);
  float* ga = gnf_a;
  float* gb = gnf_b;
  for (int t = TT - 1; t >= 0; --t) {
    const float* U = W_up + (size_t)t * 4 * 4096;
    const float* R1 = W_rad1 + (size_t)t * 512;
    const float* R2 = W_rad2 + (size_t)t * 4096;
    const float* O = W_out + (size_t)t * 4 * 4096;
    const float* S = W_skip + (size_t)t * 4 * 4096;

    dim3 mg(cdiv(NN, 128), 16);
    mix_bwd_kernel<<<mg, TB, 0, stream>>>(ga, O, S, gagg, gb, NN);
    // gsh += sum_k gagg[recv] * q
    gsh_kernel<<<cdiv((size_t)EE * 16, TB), TB, 0, stream>>>(gagg, q[t], recv,
                                                             gsh);
    // gq -> grad_rad (wrt rad), gh scatter (wrt h)
    zero_kernel<<<cdiv((size_t)NN * 64, TB), TB, 0, stream>>>(gh,
                                                              (size_t)NN * 64);
    gq_kernel<<<cdiv((size_t)EE * 64, TB), TB, 0, stream>>>(
        gagg, sh, h[t], rad[t], send, recv, grad_rad, gh);
    // gs = grad_rad @ R2^T ; gpre = gs * silu'(pre)
    gemm64_kernel<true, false><<<cdiv(EE, 128), TB, 0, stream>>>(
        grad_rad, 64, R2, gs, 64, EE);
    gpre_kernel<<<cdiv((size_t)EE * 64, TB), TB, 0, stream>>>(pre[t], gs);
    // gef += gpre @ R1^T
    gef_kernel<<<cdiv((size_t)EE * 8, TB), TB, 0, stream>>>(gs, R1, gef);
    // gnf_prev[:,0,:] += gh @ U0^T
    gemm64_kernel<true, true><<<cdiv(NN, 128), TB, 0, stream>>>(gh, 64, U, gb,
                                                                1024, NN);
    // readout t=0 feeds nf1
    if (t == 1)
      addread_kernel<<<cdiv((size_t)NN * 64, TB), TB, 0, stream>>>(W_read, gb);
    float* tmp = ga;
    ga = gb;
    gb = tmp;
  }
  // accumulate forces from gsh / gef through edge geometry
  force_kernel<<<cdiv(EE, TB), TB, 0, stream>>>(gsh, gef, vec, r, send, recv,
                                                forces);
}